// ConvPoolBlock_47519518163432
// MI455X (gfx1250) — compile-verified
//
#include <hip/hip_runtime.h>
#include <hip/hip_bf16.h>
#include <cstdint>

#define C_DIM 128
#define TPB 256
#define WPAD 130  // padded row stride for transposed W in LDS (bank-conflict free)

typedef float v2f __attribute__((ext_vector_type(2)));
typedef float v8f __attribute__((ext_vector_type(8)));

static __device__ __forceinline__ void atomicMaxF(float* addr, float val) {
  int old = __float_as_int(*addr);
  while (val > __int_as_float(old)) {
    int assumed = old;
    old = atomicCAS((int*)addr, assumed, __float_as_int(val));
    if (old == assumed) break;
  }
}

// ---------------- elementwise / init helpers ----------------
__global__ void k_fill_i32(int* p, int v, int n) {
  int i = blockIdx.x * blockDim.x + threadIdx.x;
  if (i < n) p[i] = v;
}
__global__ void k_fill_f32(float* p, float v, int n) {
  int i = blockIdx.x * blockDim.x + threadIdx.x;
  if (i < n) p[i] = v;
}
__global__ void k_deg_count(const int* __restrict__ dst, int* deg, int E_) {
  int i = blockIdx.x * blockDim.x + threadIdx.x;
  if (i < E_) atomicAdd(&deg[dst[i]], 1);
}
__global__ void k_calc_dinv(const int* __restrict__ deg, float* dinv, int n) {
  int i = blockIdx.x * blockDim.x + threadIdx.x;
  if (i < n) dinv[i] = rsqrtf((float)(deg[i] + 1));  // +1 self loop
}
// s = gamma*rsqrt(var+eps), t = beta - mean*s  (so bn(x) = x*s + t)
__global__ void k_bn_coeff(const float* __restrict__ g, const float* __restrict__ b,
                           const float* __restrict__ m, const float* __restrict__ v,
                           float* s, float* t) {
  int c = threadIdx.x;
  if (c < C_DIM) {
    float sc = g[c] * rsqrtf(v[c] + 1e-5f);
    s[c] = sc;
    t[c] = b[c] - m[c] * sc;
  }
}

// ---------------- WMMA GEMM: Y = bn(X) @ W  (N x 128 @ 128 x 128, f32) ----------------
// W is staged TRANSPOSED in LDS with padded stride so each B fragment is one
// contiguous 8B ds_load_b64 into an even-aligned VGPR pair (no repack movs).
__global__ __launch_bounds__(TPB) void k_gemm_bn_wmma(
    const float* __restrict__ xin, const float* __restrict__ w,
    const float* __restrict__ s, const float* __restrict__ t,
    float* __restrict__ y, int nrows) {
  __shared__ float sWT[C_DIM * WPAD];  // ~65KB of the WGP's 320KB LDS
  __shared__ float sS[C_DIM];
  __shared__ float sT[C_DIM];
  int tid = threadIdx.x;
  for (int i = tid; i < C_DIM * C_DIM; i += TPB) {
    int k = i >> 7, n = i & (C_DIM - 1);
    sWT[n * WPAD + k] = w[i];  // coalesced global read, conflict-free LDS write
  }
  if (tid < C_DIM) {
    sS[tid] = s[tid];
    sT[tid] = t[tid];
  }
  __syncthreads();

  int wave = tid >> 5;
  int lane = tid & 31;
  int tileRow = blockIdx.x * (TPB / 32) + wave;
  if (tileRow * 16 >= nrows) return;  // wave-uniform: EXEC stays all-ones for WMMA
  int row0 = tileRow * 16;
  int m = lane & 15;
  int hi = lane >> 4;  // lane half: K pair select for A, K pair select for B
  const float* xr = xin + (size_t)(row0 + m) * C_DIM;
  int n = lane & 15;

  v8f acc[8] = {};

  // D(16x16) += A(16x4,f32) * B(4x16,f32); K stepped by 4; C=128 -> 32 steps.
  // A layout (ISA 7.12.2): lanes 0-15 K={0,1}, lanes 16-31 K={2,3}; reg = K%2.
  // B assumed mirrored: lane = n + 16*(K/2), reg = K%2.
  for (int k0 = 0; k0 < C_DIM; k0 += 4) {
    int ka = k0 + 2 * hi;  // even -> 8B-aligned v2f loads
    v2f xa = *(const v2f*)(xr + ka);
    v2f sa = *(const v2f*)(sS + ka);
    v2f ta = *(const v2f*)(sT + ka);
    v2f a = xa * sa + ta;  // fused BatchNorm on the A operand (v_pk_fma_f32)
#pragma unroll
    for (int nb = 0; nb < 8; ++nb) {
      v2f b = *(const v2f*)(sWT + (size_t)(nb * 16 + n) * WPAD + ka);
      acc[nb] = __builtin_amdgcn_wmma_f32_16x16x4_f32(
          false, a, false, b, (short)0, acc[nb], false, false);
    }
  }

  // C/D layout: reg r -> M=r (lanes 0-15) or M=r+8 (lanes 16-31), N = lane&15
  int outM = hi * 8;
#pragma unroll
  for (int nb = 0; nb < 8; ++nb)
#pragma unroll
    for (int r = 0; r < 8; ++r)
      y[(size_t)(row0 + r + outM) * C_DIM + nb * 16 + n] = acc[nb][r];
}

// ---------------- GCN aggregation ----------------
__global__ void k_agg_init(const float* __restrict__ y, const float* __restrict__ dinv,
                           float* z, int n) {  // n = N*C ; self-loop term
  int i = blockIdx.x * blockDim.x + threadIdx.x;
  if (i < n) {
    float d = dinv[i >> 7];
    z[i] = y[i] * d * d;
  }
}
__global__ void k_agg_edge(const float* __restrict__ y, const float* __restrict__ dinv,
                           const int* __restrict__ src, const int* __restrict__ dst,
                           float* z, int E_) {
  int gid = blockIdx.x * blockDim.x + threadIdx.x;
  int j = gid >> 5;  // 32 threads per edge, 4 channels each
  if (j >= E_) return;
  int c0 = (gid & 31) * 4;
  int u = src[j], v = dst[j];
  float nrm = dinv[u] * dinv[v];
  float4 yv = *(const float4*)(y + (size_t)u * C_DIM + c0);
  float* zp = z + (size_t)v * C_DIM + c0;
  atomicAdd(zp + 0, yv.x * nrm);
  atomicAdd(zp + 1, yv.y * nrm);
  atomicAdd(zp + 2, yv.z * nrm);
  atomicAdd(zp + 3, yv.w * nrm);
}
__global__ void k_agg_fin(const float* __restrict__ z, const float* __restrict__ bias,
                          float* x, int n) {
  int i = blockIdx.x * blockDim.x + threadIdx.x;
  if (i < n) x[i] = fmaxf(z[i] + bias[i & (C_DIM - 1)], 0.0f);
}
__global__ void k_bn_apply(const float* __restrict__ x, const float* __restrict__ s,
                           const float* __restrict__ t, float* y, int n) {
  int i = blockIdx.x * blockDim.x + threadIdx.x;
  if (i < n) {
    int c = i & (C_DIM - 1);
    y[i] = x[i] * s[c] + t[c];
  }
}

// ---------------- edge pooling ----------------
__global__ void k_edge_score(const float* __restrict__ xp, const int* __restrict__ src,
                             const int* __restrict__ dst, const float* __restrict__ pw,
                             const float* __restrict__ pb, float* e, int E_) {
  int j = blockIdx.x * (TPB / 32) + (threadIdx.x >> 5);  // one wave32 per edge
  int lane = threadIdx.x & 31;
  if (j >= E_) return;
  int u = src[j], v = dst[j];
  float acc = 0.f;
#pragma unroll
  for (int c = lane; c < C_DIM; c += 32) {
    acc += xp[(size_t)u * C_DIM + c] * pw[c];
    acc += xp[(size_t)v * C_DIM + c] * pw[C_DIM + c];
  }
#pragma unroll
  for (int off = 16; off > 0; off >>= 1) acc += __shfl_xor(acc, off, 32);
  if (lane == 0) e[j] = acc + pb[0];
}
__global__ void k_segmax(const float* __restrict__ e, const int* __restrict__ dst,
                         float* fm, int E_) {
  int j = blockIdx.x * blockDim.x + threadIdx.x;
  if (j < E_) atomicMaxF(&fm[dst[j]], e[j]);
}
__global__ void k_segexp(float* e, const float* __restrict__ fm,
                         const int* __restrict__ dst, float* denom, int E_) {
  int j = blockIdx.x * blockDim.x + threadIdx.x;
  if (j < E_) {
    float ee = expf(e[j] - fm[dst[j]]);
    e[j] = ee;
    atomicAdd(&denom[dst[j]], ee);
  }
}
__global__ void k_score_fin(const float* __restrict__ e, const float* __restrict__ denom,
                            const int* __restrict__ dst, float* score, int E_) {
  int j = blockIdx.x * blockDim.x + threadIdx.x;
  if (j < E_) score[j] = e[j] / denom[dst[j]] + 0.5f;  // ADD_TO_EDGE_SCORE
}
// key = (~monotone(score) << 32) | edge_idx  -> ascending sort == descending score, stable
__global__ void k_build_keys(const float* __restrict__ score, unsigned long long* keys,
                             int E_, int sn) {
  int i = blockIdx.x * blockDim.x + threadIdx.x;
  if (i >= sn) return;
  if (i < E_) {
    unsigned bits = __float_as_uint(score[i]);
    unsigned mono = (bits & 0x80000000u) ? ~bits : (bits | 0x80000000u);
    keys[i] = ((unsigned long long)(~mono) << 32) | (unsigned)i;
  } else {
    keys[i] = ~0ull;  // padding sorts last
  }
}
__global__ void k_bitonic(unsigned long long* keys, int j, int k) {
  unsigned i = blockIdx.x * blockDim.x + threadIdx.x;
  unsigned ixj = i ^ (unsigned)j;
  if (ixj > i) {
    unsigned long long a = keys[i], b = keys[ixj];
    bool up = ((i & (unsigned)k) == 0);
    if ((a > b) == up) {
      keys[i] = b;
      keys[ixj] = a;
    }
  }
}
// Greedy matching in score order is inherently sequential -> single thread, like PyG.
__global__ void k_serial_match(const unsigned long long* __restrict__ keys,
                               const int* __restrict__ src, const int* __restrict__ dst,
                               const float* __restrict__ score, int* matched,
                               int* cluster, float* cscore, int E_, int n) {
  if (threadIdx.x != 0 || blockIdx.x != 0) return;
  int count = 0;
  for (int t = 0; t < E_; ++t) {
    int j = (int)(keys[t] & 0xFFFFFFFFull);
    int u = src[j], v = dst[j];
    if (!matched[u] && !matched[v]) {
      matched[u] = 1;
      matched[v] = 1;
      cluster[u] = count;
      cluster[v] = count;
      cscore[count] = score[j];
      ++count;
    }
  }
  for (int v = 0; v < n; ++v) {
    if (!matched[v]) {
      cluster[v] = count;
      cscore[count] = 1.0f;
      ++count;
    }
  }
}
__global__ void k_scatter_newx(const float* __restrict__ xp, const int* __restrict__ cluster,
                               float* accum, int n) {  // n = N*C
  int i = blockIdx.x * blockDim.x + threadIdx.x;
  if (i < n) {
    int v = i >> 7, c = i & (C_DIM - 1);
    atomicAdd(&accum[(size_t)cluster[v] * C_DIM + c], xp[i]);
  }
}
__global__ void k_seg_batchmax(const int* __restrict__ batch, const int* __restrict__ cluster,
                               int* ib, int n) {
  int v = blockIdx.x * blockDim.x + threadIdx.x;
  if (v < n) atomicMax(&ib[cluster[v]], batch[v]);
}
__global__ void k_out_x(const float* __restrict__ accum, const float* __restrict__ cscore,
                        float* out, int n) {
  int i = blockIdx.x * blockDim.x + threadIdx.x;
  if (i < n) out[i] = accum[i] * cscore[i >> 7];
}
__global__ void k_out_ei(const int* __restrict__ cluster, const int* __restrict__ src,
                         const int* __restrict__ dst, float* out, int E_) {
  int j = blockIdx.x * blockDim.x + threadIdx.x;
  if (j < E_) {
    out[j] = (float)cluster[src[j]];
    out[E_ + j] = (float)cluster[dst[j]];
  }
}
__global__ void k_out_batch(const int* __restrict__ ib, float* out, int n) {
  int v = blockIdx.x * blockDim.x + threadIdx.x;
  if (v < n) out[v] = (float)(ib[v] > 0 ? ib[v] : 0);
}

static inline int cdiv(long long a, long long b) { return (int)((a + b - 1) / b); }

extern "C" void kernel_launch(void* const* d_in, const int* in_sizes, int n_in,
                              void* d_out, int out_size, void* d_ws, size_t ws_size,
                              hipStream_t stream) {
  const float* x = (const float*)d_in[0];
  const int* ei = (const int*)d_in[1];
  const int* batch = (const int*)d_in[2];
  const float* bn_g = (const float*)d_in[3];
  const float* bn_b = (const float*)d_in[4];
  const float* bn_m = (const float*)d_in[5];
  const float* bn_v = (const float*)d_in[6];
  const float* conv_w = (const float*)d_in[7];
  const float* conv_b = (const float*)d_in[8];
  const float* pbn_g = (const float*)d_in[9];
  const float* pbn_b = (const float*)d_in[10];
  const float* pbn_m = (const float*)d_in[11];
  const float* pbn_v = (const float*)d_in[12];
  const float* pool_w = (const float*)d_in[13];
  const float* pool_b = (const float*)d_in[14];

  const int N = in_sizes[2];
  const int E = in_sizes[1] / 2;
  const int L = in_sizes[8] / C_DIM;
  const int NC = N * C_DIM;
  const int* src = ei;
  const int* dst = ei + E;
  int SN = 1;
  while (SN < E) SN <<= 1;  // bitonic size (next pow2)

  // workspace layout (f32 slots; even count before keys -> 8B aligned)
  float* bufA = (float*)d_ws;            // N*C  (x / Z / accum)
  float* bufB = bufA + (size_t)NC;       // N*C  (GEMM out / xp)
  float* dinv = bufB + (size_t)NC;       // N
  float* fm = dinv + N;                  // N
  float* denom = fm + N;                 // N
  float* e = denom + N;                  // E
  float* score = e + E;                  // E
  float* cscore = score + E;             // N
  float* sbn = cscore + N;               // C
  float* tbn = sbn + C_DIM;              // C
  int* itmp = (int*)(tbn + C_DIM);       // N (deg, later batch-max)
  int* matched = itmp + N;               // N
  int* cluster = matched + N;            // N
  unsigned long long* keys = (unsigned long long*)(cluster + N);  // SN
  (void)ws_size; (void)n_in; (void)out_size;

  // degrees -> dinv
  k_fill_i32<<<cdiv(N, TPB), TPB, 0, stream>>>(itmp, 0, N);
  k_deg_count<<<cdiv(E, TPB), TPB, 0, stream>>>(dst, itmp, E);
  k_calc_dinv<<<cdiv(N, TPB), TPB, 0, stream>>>(itmp, dinv, N);

  // conv layers: Y = bn(x)@W (WMMA) ; Z = scatter(norm*Y) ; x = relu(Z+b)
  const float* xin = x;
  int gemmBlocks = cdiv(N / 16, TPB / 32);
  for (int l = 0; l < L; ++l) {
    k_bn_coeff<<<1, C_DIM, 0, stream>>>(bn_g + l * C_DIM, bn_b + l * C_DIM,
                                        bn_m + l * C_DIM, bn_v + l * C_DIM, sbn, tbn);
    k_gemm_bn_wmma<<<gemmBlocks, TPB, 0, stream>>>(
        xin, conv_w + (size_t)l * C_DIM * C_DIM, sbn, tbn, bufB, N);
    k_agg_init<<<cdiv(NC, TPB), TPB, 0, stream>>>(bufB, dinv, bufA, NC);
    k_agg_edge<<<cdiv((long long)E * 32, TPB), TPB, 0, stream>>>(bufB, dinv, src, dst, bufA, E);
    k_agg_fin<<<cdiv(NC, TPB), TPB, 0, stream>>>(bufA, conv_b + l * C_DIM, bufA, NC);
    xin = bufA;
  }

  // pool batchnorm -> xp (bufB)
  k_bn_coeff<<<1, C_DIM, 0, stream>>>(pbn_g, pbn_b, pbn_m, pbn_v, sbn, tbn);
  k_bn_apply<<<cdiv(NC, TPB), TPB, 0, stream>>>(bufA, sbn, tbn, bufB, NC);

  // edge scores + softmax grouped by dst
  k_edge_score<<<cdiv(E, TPB / 32), TPB, 0, stream>>>(bufB, src, dst, pool_w, pool_b, e, E);
  k_fill_f32<<<cdiv(N, TPB), TPB, 0, stream>>>(fm, -3.402823466e38f, N);
  k_fill_f32<<<cdiv(N, TPB), TPB, 0, stream>>>(denom, 0.0f, N);
  k_fill_f32<<<cdiv(N, TPB), TPB, 0, stream>>>(cscore, 1.0f, N);
  k_segmax<<<cdiv(E, TPB), TPB, 0, stream>>>(e, dst, fm, E);
  k_segexp<<<cdiv(E, TPB), TPB, 0, stream>>>(e, fm, dst, denom, E);
  k_score_fin<<<cdiv(E, TPB), TPB, 0, stream>>>(e, denom, dst, score, E);

  // stable descending sort by score (bitonic on packed u64 keys)
  k_build_keys<<<cdiv(SN, TPB), TPB, 0, stream>>>(score, keys, E, SN);
  for (int k = 2; k <= SN; k <<= 1)
    for (int j = k >> 1; j > 0; j >>= 1)
      k_bitonic<<<cdiv(SN, TPB), TPB, 0, stream>>>(keys, j, k);

  // greedy contraction (sequential by construction)
  k_fill_i32<<<cdiv(N, TPB), TPB, 0, stream>>>(matched, 0, N);
  k_serial_match<<<1, 1, 0, stream>>>(keys, src, dst, score, matched, cluster, cscore, E, N);

  // new_x = segment_sum(xp, cluster) * cscore ; new_batch = max(segment_max(batch),0)
  k_fill_f32<<<cdiv(NC, TPB), TPB, 0, stream>>>(bufA, 0.0f, NC);
  k_scatter_newx<<<cdiv(NC, TPB), TPB, 0, stream>>>(bufB, cluster, bufA, NC);
  k_fill_i32<<<cdiv(N, TPB), TPB, 0, stream>>>(itmp, (int)0x80000000, N);
  k_seg_batchmax<<<cdiv(N, TPB), TPB, 0, stream>>>(batch, cluster, itmp, N);

  float* out = (float*)d_out;
  k_out_x<<<cdiv(NC, TPB), TPB, 0, stream>>>(bufA, cscore, out, NC);
  k_out_ei<<<cdiv(E, TPB), TPB, 0, stream>>>(cluster, src, dst, out + (size_t)NC, E);
  k_out_batch<<<cdiv(N, TPB), TPB, 0, stream>>>(itmp, out + (size_t)NC + 2 * (size_t)E, N);
}